// PointsRasterizer_51488067944949
// MI455X (gfx1250) — compile-verified
//
#include <hip/hip_runtime.h>
#include <hip/hip_bf16.h>

#define Bc 2
#define Pc 20000
#define Hc 128
#define Wc 128
#define Kc 8
#define RAD2 0.0025f
#define ZSENT 3.0e38f

// per-image staged blob in d_ws: [ float2 xy[Pc] | float z[Pc] ]  = 240000 B
#define IMG_BYTES ((size_t)Pc * 12)
#define IMG_DWORDS (Pc * 3)

typedef float v2f __attribute__((ext_vector_type(2)));
typedef float v8f __attribute__((ext_vector_type(8)));
typedef unsigned int v4u __attribute__((ext_vector_type(4)));
typedef int v4i __attribute__((ext_vector_type(4)));
typedef int v8i __attribute__((ext_vector_type(8)));

// ---------------------------------------------------------------------------
// Prep: world -> view (row-vector * R + T), perspective divide -> NDC xy, z
// ---------------------------------------------------------------------------
__global__ void pr_prep_kernel(const float* __restrict__ pts,
                               const float* __restrict__ R,
                               const float* __restrict__ T,
                               const float* __restrict__ focal,
                               char* __restrict__ ws) {
    int i = blockIdx.x * blockDim.x + threadIdx.x;
    if (i >= Bc * Pc) return;
    int b = i / Pc;
    int p = i - b * Pc;
    const float* pt = pts + (size_t)i * 3;
    const float* Rb = R + b * 9;
    const float* Tb = T + b * 3;
    float x = pt[0], y = pt[1], zz = pt[2];
    float vx = x * Rb[0] + y * Rb[3] + zz * Rb[6] + Tb[0];
    float vy = x * Rb[1] + y * Rb[4] + zz * Rb[7] + Tb[1];
    float vz = x * Rb[2] + y * Rb[5] + zz * Rb[8] + Tb[2];
    float s = focal[0] / vz;
    float2* xy = (float2*)(ws + (size_t)b * IMG_BYTES);
    float*  z  = (float*)(ws + (size_t)b * IMG_BYTES + (size_t)Pc * 8);
    xy[p] = make_float2(vx * s, vy * s);
    z[p] = vz;
}

// lane <-> lane^16 exchange via ds_bpermute (wave32: index bits [6:2])
__device__ __forceinline__ float xwaveF(float v, int lane) {
    return __int_as_float(
        __builtin_amdgcn_ds_bpermute((lane ^ 16) << 2, __float_as_int(v)));
}
__device__ __forceinline__ int xwaveI(int v, int lane) {
    return __builtin_amdgcn_ds_bpermute((lane ^ 16) << 2, v);
}

// sorted insert (ascending z) into register-resident top-8
__device__ __forceinline__ void ins8(float (&kz)[8], float (&kd)[8], int (&ki)[8],
                                     float z, float d2, int idx) {
#pragma unroll
    for (int s = 7; s >= 0; --s) {
        bool lt = z < kz[s];
        bool gep = (s == 0) ? true : (z >= kz[s - 1]);
        float nz = gep ? z  : kz[s - 1];
        float nd = gep ? d2 : kd[s - 1];
        int   ni = gep ? idx : ki[s - 1];
        kz[s] = lt ? nz : kz[s];
        kd[s] = lt ? nd : kd[s];
        ki[s] = lt ? ni : ki[s];
    }
}

// ---------------------------------------------------------------------------
// TDM stage: one wave issues tensor_load_to_lds for the whole 240000-byte
// per-image point blob (1-D tensor, 60000 dwords, data_size=4B).
// D# layout per CDNA5 ISA ch.8 (group0/group1 bitfields).
// ---------------------------------------------------------------------------
__device__ __forceinline__ void tdm_stage(const char* src, unsigned ldsAddr) {
    unsigned long long ga = (unsigned long long)(uintptr_t)src;
    v4u g0;
    g0[0] = 1u;                                    // count=1, user mode
    g0[1] = ldsAddr;                               // lds_addr
    g0[2] = (unsigned)ga;                          // global_addr[31:0]
    g0[3] = ((unsigned)(ga >> 32) & 0x01FFFFFFu)   // global_addr[56:32]
            | (2u << 30);                          // type=2 ("image")
    v8i g1;
    unsigned td0 = IMG_DWORDS;                     // 60000 dwords
    g1[0] = (int)(2u << 16);                       // data_size=4B
    g1[1] = (int)((td0 & 0xFFFFu) << 16);          // tensor_dim0[15:0]
    g1[2] = (int)((td0 >> 16) | (1u << 16));       // tensor_dim0[31:16] | tensor_dim1=1
    g1[3] = (int)((td0 & 0xFFFFu) << 16);          // tile_dim0 = 60000
    g1[4] = 0;                                     // tile_dim1=0, tile_dim2=0
    g1[5] = (int)td0;                              // tensor_dim0_stride[31:0]
    g1[6] = 0;
    g1[7] = 0;
    v4i gz = {0, 0, 0, 0};
#if __has_include(<hip/amd_detail/amd_gfx1250_TDM.h>)
    v8i gz8 = {0, 0, 0, 0, 0, 0, 0, 0};
    __builtin_amdgcn_tensor_load_to_lds(g0, g1, gz, gz, gz8, 0); // clang-23 form
#else
    __builtin_amdgcn_tensor_load_to_lds(g0, g1, gz, gz, 0);      // ROCm 7.2 form
#endif
    __builtin_amdgcn_s_wait_tensorcnt(0);
}

// ---------------------------------------------------------------------------
// Raster: one block per (b, row). 8 waves x 16 pixels = 128-pixel row.
// Whole image's point set staged in LDS (240 KB <= 320 KB WGP LDS).
// Inner loop: 2 chunks (32 points) per iteration -> two back-to-back
// V_WMMA_F32_16X16X4_F32 d^2 tiles with next chunks' A-operand preloaded.
// ---------------------------------------------------------------------------
__global__ __launch_bounds__(256, 1)
void pr_raster_kernel(const char* __restrict__ wsg, float* __restrict__ outF) {
    extern __shared__ char smem[];
    float2* sxy = (float2*)smem;                       // Pc * 8  bytes
    const float* sz = (const float*)(smem + (size_t)Pc * 8);

    const int b = blockIdx.x / Hc;
    const int h = blockIdx.x % Hc;
    const char* src = wsg + (size_t)b * IMG_BYTES;

    // TDM async stage (wave 0 issues the DMA descriptor)
    if (threadIdx.x < 32) {
        tdm_stage(src, (unsigned)(uintptr_t)smem);
    }
    __syncthreads();
    // correctness hedge: rewrite the same bytes with plain vector loads
    // (cost is negligible vs. the 625-iteration WMMA loop)
    {
        const float4* g4 = (const float4*)src;
        float4* s4 = (float4*)smem;
        const int n4 = (int)(IMG_BYTES / 16);          // 15000 float4
        for (int i = threadIdx.x; i < n4; i += 256) s4[i] = g4[i];
    }
    __syncthreads();

    const int lane = threadIdx.x & 31;
    const int wave = threadIdx.x >> 5;
    const int n  = lane & 15;        // pixel within wave's 16-pixel tile
    const int hi = lane >> 4;        // half-wave: points M=0..7 vs M=8..15
    const int w0 = wave * 16;

    // B operand (pixels), fixed for the whole loop.
    // B(0,n)=Xn  B(1,n)=Yn  B(2,n)=1  B(3,n)=Xn^2+Yn^2
    const float X  = 1.0f - (2.0f * (float)(w0 + n) + 1.0f) / (float)Wc;
    const float Yv = 1.0f - (2.0f * (float)h + 1.0f) / (float)Hc;
    v2f bm;
    bm[0] = hi ? 1.0f : X;
    bm[1] = hi ? (X * X + Yv * Yv) : Yv;

    float kz[8], kd[8];
    int ki[8];
#pragma unroll
    for (int k = 0; k < 8; ++k) { kz[k] = ZSENT; kd[k] = -1.0f; ki[k] = -1; }

    // candidate filter + rare sorted insert for one 16x16 d^2 tile
    auto process = [&](const v8f& c, int mbase) {
        float dmin = fminf(fminf(fminf(c[0], c[1]), fminf(c[2], c[3])),
                           fminf(fminf(c[4], c[5]), fminf(c[6], c[7])));
        if (dmin < RAD2) {                       // ~0.2% hit rate
            const float* zp = sz + mbase + 8 * hi;
            float zs[8];
#pragma unroll
            for (int j = 0; j < 8; ++j) zs[j] = zp[j];
#pragma unroll
            for (int j = 0; j < 8; ++j) {
                float d2 = c[j];
                float zj = zs[j];
                if (d2 < RAD2 && zj > 0.0f) {
                    ins8(kz, kd, ki, zj, d2, mbase + 8 * hi + j);
                }
            }
        }
    };

    // software-pipelined loop: 32 points (2 WMMA) per iteration
    float2 p0 = sxy[n];
    float2 p1 = sxy[16 + n];
    for (int m0 = 0; m0 < Pc; m0 += 32) {
        int mn = (m0 + 32 < Pc) ? (m0 + 32) : 0;   // wrap; last preload unused
        float2 q0 = sxy[mn + n];
        float2 q1 = sxy[mn + 16 + n];

        // A(m,0)=-2x  A(m,1)=-2y  A(m,2)=x^2+y^2  A(m,3)=1
        v2f a0, a1;
        a0[0] = hi ? (p0.x * p0.x + p0.y * p0.y) : (-2.0f * p0.x);
        a0[1] = hi ? 1.0f : (-2.0f * p0.y);
        a1[0] = hi ? (p1.x * p1.x + p1.y * p1.y) : (-2.0f * p1.x);
        a1[1] = hi ? 1.0f : (-2.0f * p1.y);

        v8f c0 = {}, c1 = {};
        c0 = __builtin_amdgcn_wmma_f32_16x16x4_f32(false, a0, false, bm,
                                                   (short)0, c0, false, false);
        c1 = __builtin_amdgcn_wmma_f32_16x16x4_f32(false, a1, false, bm,
                                                   (short)0, c1, false, false);
        process(c0, m0);
        process(c1, m0 + 16);
        p0 = q0;
        p1 = q1;
    }

    // merge the two half-wave lists for each pixel via ds_bpermute
#pragma unroll
    for (int t = 0; t < 8; ++t) {
        float oz = xwaveF(kz[t], lane);
        float od = xwaveF(kd[t], lane);
        int   oi = xwaveI(ki[t], lane);
        ins8(kz, kd, ki, oz, od, oi);
    }

    if (hi == 0) {
        const size_t NPIX = (size_t)Bc * Hc * Wc * Kc;
        size_t pix  = ((size_t)b * Hc + h) * Wc + (w0 + n);
        size_t base = pix * Kc;
        int*   idxO = (int*)outF;
        float* zO   = outF + NPIX;
        float* dO   = zO + NPIX;
#pragma unroll
        for (int k = 0; k < 8; ++k) {
            bool sel = ki[k] >= 0;
            idxO[base + k] = ki[k];
            zO[base + k]   = sel ? kz[k] : -1.0f;
            dO[base + k]   = kd[k];      // already -1 for empty slots
        }
    }
}

extern "C" void kernel_launch(void* const* d_in, const int* in_sizes, int n_in,
                              void* d_out, int out_size, void* d_ws, size_t ws_size,
                              hipStream_t stream) {
    const float* points = (const float*)d_in[0];   // [B,P,3]
    const float* R      = (const float*)d_in[1];   // [B,3,3]
    const float* T      = (const float*)d_in[2];   // [B,3]
    const float* focal  = (const float*)d_in[3];   // [1]

    pr_prep_kernel<<<(Bc * Pc + 255) / 256, 256, 0, stream>>>(
        points, R, T, focal, (char*)d_ws);

    size_t ldsBytes = IMG_BYTES;                   // 240000 B
    pr_raster_kernel<<<Bc * Hc, 256, ldsBytes, stream>>>(
        (const char*)d_ws, (float*)d_out);
}